// GraphConvolutionLayer_45801531244895
// MI455X (gfx1250) — compile-verified
//
#include <hip/hip_runtime.h>

typedef __attribute__((ext_vector_type(2))) float v2f;
typedef __attribute__((ext_vector_type(8))) float v8f;

#define FDIM 64

// ---------------- degree pipeline ----------------

__global__ void gcn_init_deg(unsigned* __restrict__ deg, int n) {
    int i = blockIdx.x * blockDim.x + threadIdx.x;
    if (i < n) deg[i] = 1u;  // self-loop guarantees deg >= 1
}

__global__ void gcn_count_deg(const long long* __restrict__ row,
                              unsigned* __restrict__ deg, int e) {
    int i = blockIdx.x * blockDim.x + threadIdx.x;
    if (i < e) atomicAdd(&deg[(int)row[i]], 1u);
}

__global__ void gcn_inv_sqrt(const unsigned* __restrict__ deg,
                             float* __restrict__ dinv, int n) {
    int i = blockIdx.x * blockDim.x + threadIdx.x;
    if (i < n) dinv[i] = rsqrtf((float)deg[i]);
}

// ---------------- support = x @ weight via V_WMMA_F32_16X16X4_F32 ----------------
// grid.x = n/16 row tiles; 128 threads = 4 waves; wave w owns output cols [16w, 16w+16).
// Per wave: 16 chained WMMAs over K=64 (K-step 4), accumulator resident in 8 VGPRs.
__global__ __launch_bounds__(128) void gcn_gemm_wmma(const float* __restrict__ x,
                                                     const float* __restrict__ w,
                                                     float* __restrict__ support) {
    const int lane = threadIdx.x & 31;
    const int wave = threadIdx.x >> 5;
    const int row0 = blockIdx.x * 16;
    const int n0   = wave * 16;
    const int mn   = lane & 15;          // M index for A, N index for B/D
    const int kb   = (lane >> 4) * 2;    // K sub-offset for this half-wave

    const float* __restrict__ xp = x + (size_t)(row0 + mn) * FDIM + kb;        // A[mn][kb + k0 + {0,1}]
    const float* __restrict__ wp = w + (size_t)kb * FDIM + n0 + mn;            // B[kb + k0 + {0,1}][n0+mn]

    v8f acc = {};
#pragma unroll
    for (int k0 = 0; k0 < FDIM; k0 += 4) {
        v2f a, b;
        a.x = xp[k0];
        a.y = xp[k0 + 1];
        b.x = wp[(size_t)k0 * FDIM];
        b.y = wp[(size_t)(k0 + 1) * FDIM];
        acc = __builtin_amdgcn_wmma_f32_16x16x4_f32(
            /*neg_a=*/false, a, /*neg_b=*/false, b,
            /*c_mod=*/(short)0, acc, /*reuse_a=*/false, /*reuse_b=*/false);
    }

    // D layout: VGPR i of lane l = D[i + 8*(l>>4)][l&15]
    float* __restrict__ sp = support + (size_t)(row0 + 8 * (lane >> 4)) * FDIM + n0 + mn;
#pragma unroll
    for (int i = 0; i < 8; ++i) sp[(size_t)i * FDIM] = acc[i];
}

// ---------------- out init: self-loop contribution + bias (non-atomic full write) ----------------
__global__ void gcn_selfloop_bias(const float* __restrict__ support,
                                  const float* __restrict__ dinv,
                                  const float* __restrict__ bias,
                                  float* __restrict__ out, int n) {
    int gid = blockIdx.x * blockDim.x + threadIdx.x;
    int i = gid >> 6;
    int f = gid & 63;
    if (i < n) {
        float d = dinv[i];
        out[gid] = support[gid] * (d * d) + bias[f];
    }
}

// ---------------- edge scatter: out[row] += support[col] * norm ----------------
// 64 threads per edge: coalesced 256B gather + coalesced f32 atomics (L2-resident).
__global__ void gcn_edge_scatter(const float* __restrict__ support,
                                 const float* __restrict__ dinv,
                                 const long long* __restrict__ row,
                                 const long long* __restrict__ col,
                                 float* __restrict__ out, int e) {
    long long gid = (long long)blockIdx.x * blockDim.x + threadIdx.x;
    int ei = (int)(gid >> 6);
    int f  = (int)(gid & 63);
    if (ei < e) {
        int r = (int)row[ei];
        int c = (int)col[ei];
        float nrm = dinv[r] * dinv[c];
        atomicAdd(&out[(size_t)r * FDIM + f], support[(size_t)c * FDIM + f] * nrm);
    }
}

extern "C" void kernel_launch(void* const* d_in, const int* in_sizes, int n_in,
                              void* d_out, int out_size, void* d_ws, size_t ws_size,
                              hipStream_t stream) {
    const float*     x      = (const float*)d_in[0];
    const long long* edge   = (const long long*)d_in[1];  // int64 [2, E]
    const float*     weight = (const float*)d_in[2];
    const float*     bias   = (const float*)d_in[3];
    float*           out    = (float*)d_out;

    const int n = in_sizes[0] / FDIM;   // 100000 (multiple of 16)
    const int e = in_sizes[1] / 2;      // 1000000

    const long long* row = edge;
    const long long* col = edge + e;

    char* ws = (char*)d_ws;
    float*    support = (float*)ws;                                    // n*64 f32
    unsigned* deg     = (unsigned*)(ws + (size_t)n * FDIM * 4);        // n u32
    float*    dinv    = (float*)(ws + (size_t)n * FDIM * 4 + (size_t)n * 4);  // n f32

    gcn_init_deg<<<(n + 255) / 256, 256, 0, stream>>>(deg, n);
    gcn_count_deg<<<(e + 255) / 256, 256, 0, stream>>>(row, deg, e);
    gcn_inv_sqrt<<<(n + 255) / 256, 256, 0, stream>>>(deg, dinv, n);

    gcn_gemm_wmma<<<n / 16, 128, 0, stream>>>(x, weight, support);

    long long outElems = (long long)n * FDIM;
    gcn_selfloop_bias<<<(unsigned)((outElems + 255) / 256), 256, 0, stream>>>(
        support, dinv, bias, out, n);

    long long edgeThreads = (long long)e * FDIM;
    gcn_edge_scatter<<<(unsigned)((edgeThreads + 255) / 256), 256, 0, stream>>>(
        support, dinv, row, col, out, e);
}